// UnlitShader_24326694765031
// MI455X (gfx1250) — compile-verified
//
#include <hip/hip_runtime.h>
#include <cstdint>

// Problem constants (reference: N=8, H=512, W=512, K=4, D=3)
constexpr int N_ = 8, H_ = 512, W_ = 512, K_ = 4, D_ = 3;
constexpr int PIXELS = N_ * H_ * W_;             // 2,097,152
constexpr int QUADS  = PIXELS / 4;               // 524,288 (4 px / thread / iter)
constexpr int BLOCK  = 256;                      // 8 wave32 per workgroup
constexpr int ITERS  = 2;                        // double-buffered pipeline depth
constexpr int GRID   = QUADS / (BLOCK * ITERS);  // 1024 blocks

typedef float f4 __attribute__((ext_vector_type(4)));

// Truncate a generic pointer to the low 32 bits = LDS byte address
// (flat->LDS aperture mapping uses addr[31:0]; async-to-LDS VDST wants the same).
__device__ __forceinline__ uint32_t lds_addr32(const void* p) {
    return (uint32_t)(uintptr_t)p;
}

// Issue the 8 async global->LDS copies for one chunk: per lane,
// 4x b128 (bary quad, 48B stride, 16B aligned) + 4x b32 (pix_to_face, stride 16B).
__device__ __forceinline__ void issue_chunk(const int*   __restrict__ pix,
                                            const float* __restrict__ bary,
                                            int p0, int lane,
                                            f4* sb, int* sp) {
    #pragma unroll
    for (int i = 0; i < 4; ++i) {
        const uint32_t lb = lds_addr32(&sb[lane * 4 + i]);
        const void*    gb = (const void*)((const char*)bary + (size_t)(p0 + i) * 48);
        asm volatile("global_load_async_to_lds_b128 %0, %1, off th:TH_LOAD_NT"
                     :: "v"(lb), "v"(gb) : "memory");

        const uint32_t lp = lds_addr32(&sp[lane * 4 + i]);
        const void*    gp = (const void*)(pix + (size_t)(p0 + i) * K_);
        asm volatile("global_load_async_to_lds_b32 %0, %1, off th:TH_LOAD_NT"
                     :: "v"(lp), "v"(gp) : "memory");
    }
}

__global__ __launch_bounds__(BLOCK)
void unlit_shade_kernel(const int*   __restrict__ pix_to_face,   // [N,H,W,K] int32
                        const float* __restrict__ bary,          // [N,H,W,K,3] f32
                        const int*   __restrict__ faces,         // [F,3] int32
                        const float* __restrict__ verts,         // [V,3] f32
                        float*       __restrict__ out)           // [N,H,W,3] f32
{
    // Per-lane-private staging: lane t only ever touches slots [t*4, t*4+4).
    // => no cross-wave hazards, sync is per-wave s_wait_asynccnt only.
    __shared__ f4  s_bary[2][BLOCK * 4];   // 2 x 16 KB
    __shared__ int s_pix [2][BLOCK * 4];   // 2 x  4 KB

    const int lane   = threadIdx.x;
    const int tid    = blockIdx.x * BLOCK + lane;
    const int stride = GRID * BLOCK;       // quads per grid iteration

    // Prologue: start streaming chunk 0 into buffer 0.
    issue_chunk(pix_to_face, bary, (tid + 0 * stride) * 4, lane, s_bary[0], s_pix[0]);

    #pragma unroll
    for (int it = 0; it < ITERS; ++it) {
        const int buf = it & 1;
        const int q   = tid + it * stride;
        const int p0  = q * 4;

        if (it + 1 < ITERS) {
            // Kick off next chunk (8 more async ops), then wait until only
            // those 8 remain outstanding => current chunk is resident in LDS.
            issue_chunk(pix_to_face, bary, (q + stride) * 4, lane,
                        s_bary[(it + 1) & 1], s_pix[(it + 1) & 1]);
            asm volatile("s_wait_asynccnt 0x8" ::: "memory");
        } else {
            asm volatile("s_wait_asynccnt 0x0" ::: "memory");
        }

        // ---- Read this lane's staged slots from LDS ------------------------
        int f[4];
        f4  b[4];
        #pragma unroll
        for (int i = 0; i < 4; ++i) {
            f[i] = s_pix [buf][lane * 4 + i];
            b[i] = s_bary[buf][lane * 4 + i];
        }

        // ---- Dependent gathers (L2-hot tables, default temporal policy) ----
        float res[12];
        #pragma unroll
        for (int i = 0; i < 4; ++i) {
            const int   fi = f[i];
            const int   fc = fi < 0 ? 0 : fi;          // branchless background
            const float w  = fi < 0 ? 0.0f : 1.0f;

            const int v0 = faces[(size_t)fc * 3 + 0];
            const int v1 = faces[(size_t)fc * 3 + 1];
            const int v2 = faces[(size_t)fc * 3 + 2];

            const float b0 = b[i].x * w;
            const float b1 = b[i].y * w;
            const float b2 = b[i].z * w;

            const float* c0 = verts + (size_t)v0 * 3;
            const float* c1 = verts + (size_t)v1 * 3;
            const float* c2 = verts + (size_t)v2 * 3;

            #pragma unroll
            for (int d = 0; d < 3; ++d)
                res[i * 3 + d] = fmaf(b0, c0[d], fmaf(b1, c1[d], b2 * c2[d]));
        }

        // ---- Write-once output: 3 x b128 non-temporal stores ---------------
        // out offset = p0*3 floats = q*48 B -> 16B aligned.
        f4* op = (f4*)(out + (size_t)p0 * 3);
        #pragma unroll
        for (int j = 0; j < 3; ++j) {
            f4 v = { res[j * 4 + 0], res[j * 4 + 1], res[j * 4 + 2], res[j * 4 + 3] };
            __builtin_nontemporal_store(v, op + j);
        }
    }
}

extern "C" void kernel_launch(void* const* d_in, const int* in_sizes, int n_in,
                              void* d_out, int out_size, void* d_ws, size_t ws_size,
                              hipStream_t stream) {
    const int*   pix_to_face = (const int*)  d_in[0];   // [N,H,W,K]
    const float* bary        = (const float*)d_in[1];   // [N,H,W,K,3]
    const int*   faces       = (const int*)  d_in[2];   // [F,3]
    const float* verts       = (const float*)d_in[3];   // [V,3]
    float*       out         = (float*)      d_out;     // [N,H,W,3]

    unlit_shade_kernel<<<GRID, BLOCK, 0, stream>>>(pix_to_face, bary, faces, verts, out);
}